// FarthestSubsample_9723805958812
// MI455X (gfx1250) — compile-verified
//
#include <hip/hip_runtime.h>
#include <cstdint>
#include <cstddef>

// ---------------------------------------------------------------------------
// Phase 1: exact FPS. One block per batch; all coords + running min distances
// live in VGPRs (256 thr x 32 pts x {x,y,z,d}). Block argmax carries the
// winning point's coords through the wave32 shuffle tree, so each iteration is
// just: 32 fused dist/min updates + wave shfl_xor argmax + 2 barriers.
// ---------------------------------------------------------------------------
#define FPS_T   256
#define FPS_PPT 32              // points per thread; N <= FPS_T*FPS_PPT = 8192
#define FPS_NW  (FPS_T / 32)    // waves per block (wave32)

struct Cand { float v; float x, y, z; int i; };

__device__ __forceinline__ Cand cand_better(Cand a, Cand b) {
  // argmax with lowest-index tie-break (matches jnp.argmax "first max")
  bool take = (b.v > a.v) || (b.v == a.v && b.i < a.i);
  return take ? b : a;
}

__device__ __forceinline__ Cand wave_argmax(Cand c) {
#pragma unroll
  for (int off = 16; off > 0; off >>= 1) {
    Cand o;
    o.v = __shfl_xor(c.v, off, 32);
    o.x = __shfl_xor(c.x, off, 32);
    o.y = __shfl_xor(c.y, off, 32);
    o.z = __shfl_xor(c.z, off, 32);
    o.i = __shfl_xor(c.i, off, 32);
    c = cand_better(c, o);
  }
  return c;
}

__device__ __forceinline__ void block_argmax(
    Cand c, int lane, int wave,
    float* sv, float* sx, float* sy, float* sz, int* si,
    float* rx, float* ry, float* rz, int* ri) {
  c = wave_argmax(c);
  if (lane == 0) { sv[wave] = c.v; sx[wave] = c.x; sy[wave] = c.y; sz[wave] = c.z; si[wave] = c.i; }
  __syncthreads();
  if (wave == 0) {
    Cand d;
    if (lane < FPS_NW) { d.v = sv[lane]; d.x = sx[lane]; d.y = sy[lane]; d.z = sz[lane]; d.i = si[lane]; }
    else               { d.v = -1.0f; d.x = 0.f; d.y = 0.f; d.z = 0.f; d.i = 0x7fffffff; }
    d = wave_argmax(d);
    if (lane == 0) { *rx = d.x; *ry = d.y; *rz = d.z; *ri = d.i; }
  }
  __syncthreads();
}

__global__ __launch_bounds__(FPS_T) void fps_kernel(
    const float* __restrict__ coords, const int* __restrict__ init_far,
    int* __restrict__ fps_idx, int N, int S) {
  const int b = blockIdx.x;
  const int t = threadIdx.x;
  const int lane = t & 31;
  const int wave = t >> 5;

  __shared__ float sv[FPS_NW], sx[FPS_NW], sy[FPS_NW], sz[FPS_NW];
  __shared__ int   si[FPS_NW];
  __shared__ float rx, ry, rz;
  __shared__ int   ri;

  // Register-resident point set + running min distances.
  float px[FPS_PPT], py[FPS_PPT], pz[FPS_PPT], pd[FPS_PPT];
  const float* cb = coords + (size_t)b * N * 3;
#pragma unroll
  for (int k = 0; k < FPS_PPT; ++k) {
    int i = t + k * FPS_T;
    if (i < N) { px[k] = cb[3 * i + 0]; py[k] = cb[3 * i + 1]; pz[k] = cb[3 * i + 2]; }
    else       { px[k] = 0.f; py[k] = 0.f; pz[k] = 0.f; }
    pd[k] = 1e8f;  // matches torch/jax INIT_DIST
  }

  // Seed: locate init_farthest's coords via the same reduction machinery
  // (no dynamically-indexed register access anywhere).
  const int f0 = init_far[b];
  {
    Cand c; c.v = -1.0f; c.x = 0.f; c.y = 0.f; c.z = 0.f; c.i = 0x7fffffff;
#pragma unroll
    for (int k = 0; k < FPS_PPT; ++k) {
      int i = t + k * FPS_T;
      if (i == f0) { c.v = 0.f; c.x = px[k]; c.y = py[k]; c.z = pz[k]; c.i = i; }
    }
    block_argmax(c, lane, wave, sv, sx, sy, sz, si, &rx, &ry, &rz, &ri);
  }

  int* out = fps_idx + (size_t)b * S;
  for (int s = 0; s < S; ++s) {
    const float cx = rx, cy = ry, cz = rz;
    const int   ci = ri;
    if (t == 0) out[s] = ci;  // reference emits current farthest, then updates

    Cand c; c.v = -1.0f; c.x = 0.f; c.y = 0.f; c.z = 0.f; c.i = 0x7fffffff;
#pragma unroll
    for (int k = 0; k < FPS_PPT; ++k) {
      int i = t + k * FPS_T;
      float dx = px[k] - cx, dy = py[k] - cy, dz = pz[k] - cz;
      float d  = dx * dx + dy * dy + dz * dz;
      float nd = fminf(pd[k], d);
      pd[k] = nd;
      bool take = (i < N) && ((nd > c.v) || (nd == c.v && i < c.i));
      if (take) { c.v = nd; c.x = px[k]; c.y = py[k]; c.z = pz[k]; c.i = i; }
    }
    block_argmax(c, lane, wave, sv, sx, sy, sz, si, &rx, &ry, &rz, &ri);
  }
}

// ---------------------------------------------------------------------------
// Phase 2: bandwidth-bound row gather. One wave32 per output point. The 512 B
// values row moves via the CDNA5 async-tensor path: 32 lanes x b128
// global -> LDS, s_wait_asynccnt, LDS -> global, no VGPR round trip (ASYNCcnt).
// ---------------------------------------------------------------------------
#define GW 8  // waves (rows) per block

__global__ __launch_bounds__(GW * 32) void gather_kernel(
    const float* __restrict__ coords, const float* __restrict__ values,
    const float* __restrict__ mask,   const int* __restrict__ fps_idx,
    float* __restrict__ out_coords, float* __restrict__ out_values,
    float* __restrict__ out_mask, int N, int S, int C) {
  __shared__ float tile[GW * 32 * 4];  // 512 B staging per wave

  const int b    = blockIdx.y;
  const int wave = threadIdx.x >> 5;
  const int lane = threadIdx.x & 31;
  const int s    = blockIdx.x * GW + wave;
  if (s >= S) return;

  const int idx = fps_idx[(size_t)b * S + s];

  // values row: C*4 bytes, 16 B per lane per pass
  const float* src = values     + ((size_t)b * N + idx) * C;
  float*       dst = out_values + ((size_t)b * S + s)   * C;
  const int nbytes = C * 4;

  for (int off = lane * 16; off < nbytes; off += 32 * 16) {
    unsigned lds = (unsigned)(uintptr_t)(const void*)&tile[wave * 128 + (off >> 2)];
    unsigned long long ga = (unsigned long long)(uintptr_t)src + (unsigned)off;
    asm volatile("global_load_async_to_lds_b128 %0, %1, off"
                 :: "v"(lds), "v"(ga) : "memory");
  }
  asm volatile("s_wait_asynccnt 0x0" ::: "memory");
  for (int off = lane * 16; off < nbytes; off += 32 * 16) {
    unsigned lds = (unsigned)(uintptr_t)(const void*)&tile[wave * 128 + (off >> 2)];
    unsigned long long gd = (unsigned long long)(uintptr_t)dst + (unsigned)off;
    asm volatile("global_store_async_from_lds_b128 %0, %1, off"
                 :: "v"(gd), "v"(lds) : "memory");
  }
  asm volatile("s_wait_asynccnt 0x0" ::: "memory");

  // coords (3 floats) + mask (1 float): plain lane copies
  if (lane < 3)
    out_coords[((size_t)b * S + s) * 3 + lane] = coords[((size_t)b * N + idx) * 3 + lane];
  if (lane == 0)
    out_mask[(size_t)b * S + s] = mask[(size_t)b * N + idx];
}

// ---------------------------------------------------------------------------
extern "C" void kernel_launch(void* const* d_in, const int* in_sizes, int n_in,
                              void* d_out, int out_size, void* d_ws, size_t ws_size,
                              hipStream_t stream) {
  const float* coords   = (const float*)d_in[0];
  const float* values   = (const float*)d_in[1];
  const float* mask     = (const float*)d_in[2];
  const int*   init_far = (const int*)d_in[3];

  const int B = in_sizes[3];                 // [B]
  const int N = in_sizes[2] / B;             // mask is [B,N]
  const int C = in_sizes[1] / (B * N);       // values is [B,N,C]
  const int S = (int)((double)N * 0.5 + 0.5);// npoint = round(N * DS_FRAC)

  int*   fps_idx    = (int*)d_ws;            // B*S ints of scratch
  float* out_coords = (float*)d_out;
  float* out_values = out_coords + (size_t)B * S * 3;
  float* out_mask   = out_values + (size_t)B * S * C;

  fps_kernel<<<dim3(B), dim3(FPS_T), 0, stream>>>(coords, init_far, fps_idx, N, S);

  dim3 g((S + GW - 1) / GW, B);
  gather_kernel<<<g, dim3(GW * 32), 0, stream>>>(coords, values, mask, fps_idx,
                                                 out_coords, out_values, out_mask,
                                                 N, S, C);
}